// RNNTLoss_25237227831976
// MI455X (gfx1250) — compile-verified
//
#include <hip/hip_runtime.h>
#include <math.h>

#define B_  8
#define T_  256
#define U_  64
#define V_  512
#define U1  65

typedef float v2f __attribute__((ext_vector_type(2)));
typedef float v8f __attribute__((ext_vector_type(8)));

// ---------------------------------------------------------------------------
// Kernel 1: per-row log-softmax stats. One wave (32 lanes) per (b,t,u) row of
// 512 f32 logits. Row is streamed to LDS with global_load_async_to_lds_b128
// (ASYNCcnt), then reduced with shfl-xor. Emits blank_lp and emit_lp only.
// ---------------------------------------------------------------------------
__global__ __launch_bounds__(256) void k_softmax_stats(
    const float* __restrict__ logits, const int* __restrict__ targets,
    float* __restrict__ blank_lp, float* __restrict__ emit_lp) {
  __shared__ float smem[8 * V_];
  const int wave = threadIdx.x >> 5;
  const int lane = threadIdx.x & 31;
  const int r = blockIdx.x * 8 + wave;          // row id in [0, B*T*U1)
  const float* row = logits + (size_t)r * V_;

  float* sp = &smem[wave * V_];
  // low 32 bits of the flat address of an LDS object == DS byte offset
  unsigned lds0 = (unsigned)(unsigned long long)(void*)sp;

  #pragma unroll
  for (int j = 0; j < 4; ++j) {
    int e = j * 32 + lane;                       // 128 x 16B chunks per row
    unsigned long long ga = (unsigned long long)(row + e * 4);
    unsigned la = lds0 + e * 16;
    asm volatile("global_load_async_to_lds_b128 %0, %1, off"
                 :: "v"(la), "v"(ga) : "memory");
  }
  asm volatile("s_wait_asynccnt 0x0" ::: "memory");

  float x[16];
  #pragma unroll
  for (int i = 0; i < 16; ++i) x[i] = sp[i * 32 + lane];  // conflict-free

  float m = x[0];
  #pragma unroll
  for (int i = 1; i < 16; ++i) m = fmaxf(m, x[i]);
  #pragma unroll
  for (int off = 16; off >= 1; off >>= 1) m = fmaxf(m, __shfl_xor(m, off, 32));

  float s = 0.0f;
  #pragma unroll
  for (int i = 0; i < 16; ++i) s += expf(x[i] - m);
  #pragma unroll
  for (int off = 16; off >= 1; off >>= 1) s += __shfl_xor(s, off, 32);

  if (lane == 0) {
    float lse = m + logf(s);
    int b   = r / (T_ * U1);
    int rem = r % (T_ * U1);
    int t   = rem / U1;
    int u   = rem % U1;
    blank_lp[r] = sp[0] - lse;
    if (u < U_) {
      int tgt = targets[b * U_ + u];
      emit_lp[(size_t)(b * T_ + t) * U_ + u] = sp[tgt] - lse;
    }
  }
}

// ---------------------------------------------------------------------------
// Kernel 2: cum_emit via WMMA. For 16 (b,t) rows at once:
//   Y(16x64) = X(16x64) * L  with L[k][n] = (k <= n)  (upper-triangular ones)
// computed as 4 output tiles with triangular K-blocking, f32 16x16x4 WMMA.
// cum_emit[...,0] = 0; cum_emit[...,u] = inclusive_cumsum[u-1].
// ---------------------------------------------------------------------------
__global__ __launch_bounds__(32) void k_cumsum_wmma(
    const float* __restrict__ emit_lp, float* __restrict__ cum) {
  __shared__ float X[16 * U_];
  const int lane = threadIdx.x;
  const float4* s4 = (const float4*)(emit_lp + (size_t)blockIdx.x * 16 * U_);
  float4* x4 = (float4*)X;
  for (int i = lane; i < 16 * U_ / 4; i += 32) x4[i] = s4[i];
  __syncthreads();

  const int mrow = lane & 15;       // A: M = lane%16 in both halves
  const int half = lane >> 4;       // K split across lane halves
  const int nloc = lane & 15;       // B/D: N = lane%16
  const int Rbase = blockIdx.x * 16;

  #pragma unroll
  for (int j = 0; j < 4; ++j) {                  // output u-tile
    v8f acc = {};
    for (int i = 0; i <= j; ++i) {               // K-block (triangular)
      #pragma unroll
      for (int kc = 0; kc < 4; ++kc) {           // K=4 per WMMA
        int kloc = kc * 4 + half * 2;            // local k within 16-block
        v2f a;
        a.x = X[mrow * U_ + i * 16 + kloc];
        a.y = X[mrow * U_ + i * 16 + kloc + 1];
        v2f bm;                                  // L fragment, synthesized
        bm.x = (i < j || (kloc     <= nloc)) ? 1.0f : 0.0f;
        bm.y = (i < j || (kloc + 1 <= nloc)) ? 1.0f : 0.0f;
        acc = __builtin_amdgcn_wmma_f32_16x16x4_f32(
            false, a, false, bm, (short)0, acc, false, false);
      }
    }
    #pragma unroll
    for (int p = 0; p < 8; ++p) {                // D: VGPR p -> M=p(+8)
      int rowo = p + half * 8;
      cum[(size_t)(Rbase + rowo) * U1 + 1 + j * 16 + nloc] = acc[p];
    }
  }
  if (lane < 16) cum[(size_t)(Rbase + lane) * U1] = 0.0f;
}

// ---------------------------------------------------------------------------
// Kernel 3: alpha recurrence. One wave per batch; each lane owns 3 u-slots.
// Cumulative logsumexp per t-step via (m,s) monoid: local 3-elt scan +
// 5-step shfl_up wave scan. No LDS barriers on the sequential critical path.
// ---------------------------------------------------------------------------
__global__ __launch_bounds__(32) void k_alpha_scan(
    const float* __restrict__ blank_lp, const float* __restrict__ cum,
    const int* __restrict__ loglen, const int* __restrict__ tgtlen,
    float* __restrict__ costs) {
  const int b = blockIdx.x;
  const int lane = threadIdx.x;
  const float* bb = blank_lp + (size_t)b * T_ * U1;
  const float* cb = cum      + (size_t)b * T_ * U1;
  const int tidx = loglen[b] - 1;
  const int uidx = tgtlen[b];
  const int u0 = lane * 3;
  const float NEG = -1e30f;

  float alpha[3];
  #pragma unroll
  for (int jj = 0; jj < 3; ++jj) {
    int u = u0 + jj;
    alpha[jj] = (u < U1) ? cb[u] : 0.0f;       // alpha0 = cum_emit[:,0,:]
  }
  const bool owner = (u0 < U1) && (uidx >= u0) && (uidx < u0 + 3);
  const int oj = uidx - u0;
  if (tidx == 0 && owner)
    costs[b] = -(alpha[oj] + bb[(size_t)tidx * U1 + uidx]);

  for (int t = 1; t < T_; ++t) {
    float im[3], is[3], ct[3];
    float pm = NEG, ps = 0.0f;                 // lane-local running (m,s)
    #pragma unroll
    for (int jj = 0; jj < 3; ++jj) {
      int u = u0 + jj;
      bool valid = u < U1;
      float c = valid ? cb[(size_t)t * U1 + u] : 0.0f;
      ct[jj] = c;
      float f  = valid ? (alpha[jj] + bb[(size_t)(t - 1) * U1 + u] - c) : NEG;
      float fs = valid ? 1.0f : 0.0f;
      float M = fmaxf(pm, f);
      float S = ps * expf(pm - M) + fs * expf(f - M);
      pm = M; ps = S;
      im[jj] = pm; is[jj] = ps;
    }
    // inclusive wave scan of lane totals
    float tm = pm, ts = ps;
    #pragma unroll
    for (int off = 1; off < 32; off <<= 1) {
      float om = __shfl_up(tm, off, 32);
      float os = __shfl_up(ts, off, 32);
      if (lane >= off) {
        float M = fmaxf(tm, om);
        float S = ts * expf(tm - M) + os * expf(om - M);
        tm = M; ts = S;
      }
    }
    // exclusive prefix = inclusive of lane-1
    float em = __shfl_up(tm, 1, 32);
    float es = __shfl_up(ts, 1, 32);
    if (lane == 0) { em = NEG; es = 0.0f; }
    #pragma unroll
    for (int jj = 0; jj < 3; ++jj) {
      float M = fmaxf(em, im[jj]);
      float S = es * expf(em - M) + is[jj] * expf(im[jj] - M);
      if (u0 + jj < U1) alpha[jj] = ct[jj] + M + logf(S);
    }
    if (t == tidx && owner)
      costs[b] = -(alpha[oj] + bb[(size_t)tidx * U1 + uidx]);
  }
}

__global__ void k_mean(const float* __restrict__ costs, float* __restrict__ out) {
  if (threadIdx.x == 0 && blockIdx.x == 0) {
    float s = 0.0f;
    for (int i = 0; i < B_; ++i) s += costs[i];
    out[0] = s / (float)B_;
  }
}

extern "C" void kernel_launch(void* const* d_in, const int* in_sizes, int n_in,
                              void* d_out, int out_size, void* d_ws, size_t ws_size,
                              hipStream_t stream) {
  const float* logits  = (const float*)d_in[0];
  const int*   targets = (const int*)d_in[1];
  const int*   loglen  = (const int*)d_in[2];
  const int*   tgtlen  = (const int*)d_in[3];

  float* ws = (float*)d_ws;
  float* blank_lp = ws;                                   // B*T*U1 = 133120
  float* emit_lp  = blank_lp + (size_t)B_ * T_ * U1;      // B*T*U  = 131072
  float* cum      = emit_lp  + (size_t)B_ * T_ * U_;      // B*T*U1 = 133120
  float* costs    = cum      + (size_t)B_ * T_ * U1;      // B

  const int rows = B_ * T_ * U1;                          // 133120 (mult of 8)
  k_softmax_stats<<<rows / 8, 256, 0, stream>>>(logits, targets, blank_lp, emit_lp);
  k_cumsum_wmma<<<(B_ * T_) / 16, 32, 0, stream>>>(emit_lp, cum);
  k_alpha_scan<<<B_, 32, 0, stream>>>(blank_lp, cum, loglen, tgtlen, costs);
  k_mean<<<1, 1, 0, stream>>>(costs, (float*)d_out);
}